// DGN_Conv_62019327754383
// MI455X (gfx1250) — compile-verified
//
#include <hip/hip_runtime.h>
#include <hip/hip_bf16.h>

// ---------------------------------------------------------------------------
// DGN encoder + 2x graph attention for MI455X (gfx1250).
//  - GEMMs: V_WMMA_F32_16X16X4_F32 (exact fp32 on the matrix pipes),
//    A tiles staged through LDS by the Tensor Data Mover (tensor_load_to_lds,
//    TENSORcnt-synchronized, TDM pad feature for bank-conflict-free reads).
//  - Edge softmax/aggregation: float4 loads + native global f32 atomics.
// ---------------------------------------------------------------------------

typedef float v2f  __attribute__((ext_vector_type(2)));
typedef float v8f  __attribute__((ext_vector_type(8)));
typedef unsigned int u32x4 __attribute__((ext_vector_type(4)));
typedef int   i32x4 __attribute__((ext_vector_type(4)));
typedef int   i32x8 __attribute__((ext_vector_type(8)));

#define H_DIM    128
#define OUT_LD   384
#define NUM_H    8
#define HEAD_D   16

#define KC         64            // K-chunk staged per TDM transfer
#define LDS_STRIDE 68            // 64 payload dwords + 4 pad dwords per row

// ---------------------------------------------------------------------------
// TDM: DMA a [tile_rows x tile_cols] fp32 tile (row stride = row_stride_elems)
// from global memory into LDS at byte offset lds_off, padding each 64-dword
// row with 4 dwords (row stride 68 dwords in LDS -> conflict-free lane reads).
// Descriptor packing per CDNA5 ISA section 8 (D# group0 + group1, 2D tensor).
// ---------------------------------------------------------------------------
__device__ __forceinline__ void tdm_load_2d_f32(const float* gsrc,
                                                unsigned int lds_off,
                                                int tile_cols, int tile_rows,
                                                int row_stride_elems)
{
    unsigned long long ga = (unsigned long long)(size_t)gsrc;

    u32x4 g0;
    g0[0] = 1u;                                   // count=1 (valid user descriptor)
    g0[1] = lds_off;                              // lds_addr (bytes)
    g0[2] = (unsigned int)ga;                     // global_addr[31:0]
    g0[3] = (unsigned int)((ga >> 32) & 0x01FFFFFFu) | (2u << 30); // addr[56:32] | type=2

    i32x8 g1;
    //  data_size=2 (4B) | pad_enable | pad_interval=5 (64 dw) | pad_amount=3 (4 dw)
    g1[0] = (int)((2u << 16) | (1u << 20) | (5u << 22) | (3u << 25));
    // tensor_dim0 [79:48] = tile_cols ; tensor_dim1 [111:80] = tile_rows
    g1[1] = (int)(((unsigned int)tile_cols & 0xFFFFu) << 16);
    g1[2] = (int)((((unsigned int)tile_cols >> 16) & 0xFFFFu) |
                  (((unsigned int)tile_rows & 0xFFFFu) << 16));
    // tensor_dim1 hi | tile_dim0 [127:112]
    g1[3] = (int)((((unsigned int)tile_rows >> 16) & 0xFFFFu) |
                  (((unsigned int)tile_cols & 0xFFFFu) << 16));
    g1[4] = (int)((unsigned int)tile_rows & 0xFFFFu);   // tile_dim1 ; tile_dim2=0
    g1[5] = (int)(unsigned int)row_stride_elems;        // tensor_dim0_stride[31:0]
    g1[6] = 0;                                          // stride hi / dim1_stride lo
    g1[7] = 0;

    i32x4 z4 = {0, 0, 0, 0};
#if defined(__clang_major__) && (__clang_major__ >= 23)
    i32x8 z8 = {0, 0, 0, 0, 0, 0, 0, 0};
    __builtin_amdgcn_tensor_load_to_lds(g0, g1, z4, z4, z8, 0);
#else
    __builtin_amdgcn_tensor_load_to_lds(g0, g1, z4, z4, 0);
#endif
}

// ---------------------------------------------------------------------------
// fp32 WMMA GEMM: C[M,N] = act(A[M,K] @ W[K,N] + bias)
// Block = 128 threads (4 waves) -> one 16-row x 64-col output strip.
// All 4 waves share a TDM-staged 16xKC A tile in LDS; each wave owns one
// 16x16 N-tile. B fragments are coalesced global loads (W stays hot in L2).
//
// A lane layout (16x4 fp32): lanes 0-15 -> M=l, K pair {k0,k0+1};
//                            lanes 16-31 -> K pair {k0+2,k0+3}
// B lane layout (4x16 fp32): lanes 0-15 -> N=l, K pair {k0,k0+1}; 16-31 hi pair
// C lane layout: lane (half*16+l), VGPR r -> C[row0 + half*8 + r][col0 + l]
// ---------------------------------------------------------------------------
__global__ __launch_bounds__(128)
void gemm_wmma_f32_kernel(const float* __restrict__ A, int lda,
                          const float* __restrict__ W, int ldw,
                          const float* __restrict__ bias,
                          float* __restrict__ C, int ldc,
                          int K, int doRelu)
{
    __shared__ float lds_a[16 * LDS_STRIDE];

    const int row0 = blockIdx.x << 4;
    const int wave = threadIdx.x >> 5;
    const int col0 = (blockIdx.y << 6) + (wave << 4);

    const int lane = threadIdx.x & 31;
    const int half = lane >> 4;      // 0: K pair {0,1}, 1: K pair {2,3}
    const int l    = lane & 15;

    v8f acc = {0.f, 0.f, 0.f, 0.f, 0.f, 0.f, 0.f, 0.f};
    const unsigned int lds_off = (unsigned int)(size_t)(void*)lds_a;

    for (int kk = 0; kk < K; kk += KC) {
        if (wave == 0) {
            // One DMA per block: 16 rows x 64 contiguous floats, stride lda.
            tdm_load_2d_f32(A + (size_t)row0 * lda + kk, lds_off, KC, 16, lda);
            __builtin_amdgcn_s_wait_tensorcnt(0);
        }
        __syncthreads();

        const float* arow = &lds_a[l * LDS_STRIDE + 2 * half];
        const float* bcol = W + (size_t)(kk + 2 * half) * ldw + (col0 + l);
#pragma unroll 4
        for (int k0 = 0; k0 < KC; k0 += 4) {
            v2f a, b;
            a.x = arow[0];           // ds_load_b64, conflict-free (stride 68 dw)
            a.y = arow[1];
            b.x = bcol[0];           // coalesced 64B row segments of W
            b.y = bcol[ldw];
            acc = __builtin_amdgcn_wmma_f32_16x16x4_f32(
                      /*neg_a=*/false, a, /*neg_b=*/false, b,
                      /*c_mod=*/(short)0, acc, /*reuse_a=*/false, /*reuse_b=*/false);
            arow += 4;
            bcol += (size_t)4 * ldw;
        }
        __syncthreads();
    }

    const float bv  = bias[col0 + l];
    const int   row = row0 + half * 8;
#pragma unroll
    for (int r = 0; r < 8; ++r) {
        float v = acc[r] + bv;
        if (doRelu) v = fmaxf(v, 0.f);
        C[(size_t)(row + r) * ldc + (col0 + l)] = v;
    }
}

// ---------------------------------------------------------------------------
// Segment-max helper: atomic float max via int/uint monotone bit trick.
// ---------------------------------------------------------------------------
__device__ __forceinline__ void atomicMaxFloat(float* addr, float val) {
    if (val >= 0.f) {
        atomicMax((int*)addr, __float_as_int(val));
    } else {
        atomicMin((unsigned int*)addr, __float_as_uint(val));
    }
}

// init m = -inf, s = 0 over N*8 entries
__global__ void init_ms_kernel(float* __restrict__ m, float* __restrict__ s, int n)
{
    int t = blockIdx.x * blockDim.x + threadIdx.x;
    if (t < n) {
        ((unsigned int*)m)[t] = 0xFF800000u;   // -inf
        s[t] = 0.f;
    }
}

// zero out[:, colOff : colOff+128]
__global__ void zero_cols_kernel(float* __restrict__ out, int colOff, int n /* = N*128 */)
{
    int t = blockIdx.x * blockDim.x + threadIdx.x;
    if (t < n) {
        int r = t >> 7, c = t & 127;
        out[(size_t)r * OUT_LD + colOff + c] = 0.f;
    }
}

// relu in place on out[:, colOff : colOff+128]
__global__ void relu_cols_kernel(float* __restrict__ out, int colOff, int n)
{
    int t = blockIdx.x * blockDim.x + threadIdx.x;
    if (t < n) {
        int r = t >> 7, c = t & 127;
        size_t idx = (size_t)r * OUT_LD + colOff + c;
        out[idx] = fmaxf(out[idx], 0.f);
    }
}

// per-(edge, head) attention logit + segment max into m[dst][h]
__global__ void edge_score_kernel(const float* __restrict__ q,
                                  const float* __restrict__ k,
                                  const int* __restrict__ src,
                                  const int* __restrict__ dst,
                                  float* __restrict__ score,
                                  float* __restrict__ m,
                                  int n /* = E*8 */)
{
    int t = blockIdx.x * blockDim.x + threadIdx.x;
    if (t >= n) return;
    int e = t >> 3, h = t & 7;
    int d = dst[e], si = src[e];

    const float4* q4 = (const float4*)(q + (size_t)d  * H_DIM + h * HEAD_D);
    const float4* k4 = (const float4*)(k + (size_t)si * H_DIM + h * HEAD_D);
    float acc = 0.f;
#pragma unroll
    for (int i = 0; i < 4; ++i) {
        float4 a = q4[i], b = k4[i];
        acc += a.x * b.x + a.y * b.y + a.z * b.z + a.w * b.w;
    }
    acc *= 0.25f;   // 1/sqrt(HEAD_D)
    score[(size_t)e * NUM_H + h] = acc;
    atomicMaxFloat(&m[(size_t)d * NUM_H + h], acc);
}

// e = exp(score - m[dst]); segment sum into s[dst][h]; score overwritten with e
__global__ void edge_exp_kernel(float* __restrict__ score,
                                const float* __restrict__ m,
                                float* __restrict__ s,
                                const int* __restrict__ dst,
                                int n /* = E*8 */)
{
    int t = blockIdx.x * blockDim.x + threadIdx.x;
    if (t >= n) return;
    int e = t >> 3, h = t & 7;
    int d = dst[e];
    float ev = __expf(score[(size_t)e * NUM_H + h] - m[(size_t)d * NUM_H + h]);
    score[(size_t)e * NUM_H + h] = ev;
    atomicAdd(&s[(size_t)d * NUM_H + h], ev);
}

// out[dst, colOff + :] += (e / (s+1e-9)) * v[src, :]
// 32 threads per edge, 4 consecutive elements each (head h = lane>>2).
__global__ void edge_agg_kernel(const float* __restrict__ escore,
                                const float* __restrict__ s,
                                const float* __restrict__ v,
                                const int* __restrict__ src,
                                const int* __restrict__ dst,
                                float* __restrict__ out, int colOff,
                                int n /* = E*32 */)
{
    int t = blockIdx.x * blockDim.x + threadIdx.x;
    if (t >= n) return;
    int e = t >> 5, l = t & 31;
    int h = l >> 2;            // 4 lanes per head
    int c = (l & 3) * 4;       // element offset within head
    int d = dst[e], si = src[e];

    float ev    = escore[(size_t)e * NUM_H + h];
    float alpha = ev / (s[(size_t)d * NUM_H + h] + 1e-9f);

    const float4 vv = *(const float4*)(v + (size_t)si * H_DIM + h * HEAD_D + c);
    float* orow = out + (size_t)d * OUT_LD + colOff + h * HEAD_D + c;
    atomicAdd(orow + 0, alpha * vv.x);
    atomicAdd(orow + 1, alpha * vv.y);
    atomicAdd(orow + 2, alpha * vv.z);
    atomicAdd(orow + 3, alpha * vv.w);
}

// ---------------------------------------------------------------------------
// Host-side orchestration
// ---------------------------------------------------------------------------
static inline void launch_gemm(const float* A, int lda, const float* W, int ldw,
                               const float* bias, float* C, int ldc,
                               int M, int N, int K, int doRelu, hipStream_t stream)
{
    dim3 grid(M >> 4, N >> 6);   // 16-row tiles x 64-col strips (4 waves/block)
    gemm_wmma_f32_kernel<<<grid, 128, 0, stream>>>(A, lda, W, ldw, bias, C, ldc, K, doRelu);
}

static inline void run_gat_layer(const float* x, int ldx,
                                 const float* Wq, const float* bq,
                                 const float* Wk, const float* bk,
                                 const float* Wv, const float* bv,
                                 float* out, int colOff,
                                 float* q, float* kbuf, float* vbuf,
                                 float* scoreBuf, float* m, float* s,
                                 const int* src, const int* dst,
                                 int N, int E, hipStream_t stream)
{
    // QKV projections (WMMA fp32, bias, no relu)
    launch_gemm(x, ldx, Wq, H_DIM, bq, q,    H_DIM, N, H_DIM, H_DIM, 0, stream);
    launch_gemm(x, ldx, Wk, H_DIM, bk, kbuf, H_DIM, N, H_DIM, H_DIM, 0, stream);
    launch_gemm(x, ldx, Wv, H_DIM, bv, vbuf, H_DIM, N, H_DIM, H_DIM, 0, stream);

    int nms = N * NUM_H;
    init_ms_kernel<<<(nms + 255) / 256, 256, 0, stream>>>(m, s, nms);
    int ncol = N * H_DIM;
    zero_cols_kernel<<<(ncol + 255) / 256, 256, 0, stream>>>(out, colOff, ncol);

    int neh = E * NUM_H;
    edge_score_kernel<<<(neh + 255) / 256, 256, 0, stream>>>(q, kbuf, src, dst, scoreBuf, m, neh);
    edge_exp_kernel<<<(neh + 255) / 256, 256, 0, stream>>>(scoreBuf, m, s, dst, neh);

    int nagg = E * 32;
    edge_agg_kernel<<<(nagg + 255) / 256, 256, 0, stream>>>(scoreBuf, s, vbuf, src, dst,
                                                            out, colOff, nagg);
    relu_cols_kernel<<<(ncol + 255) / 256, 256, 0, stream>>>(out, colOff, ncol);
}

extern "C" void kernel_launch(void* const* d_in, const int* in_sizes, int n_in,
                              void* d_out, int out_size, void* d_ws, size_t ws_size,
                              hipStream_t stream)
{
    const float* obs = (const float*)d_in[0];
    const int*   src = (const int*)d_in[1];
    const int*   dst = (const int*)d_in[2];
    const float* W1  = (const float*)d_in[3];
    const float* b1  = (const float*)d_in[4];
    const float* W2  = (const float*)d_in[5];
    const float* b2  = (const float*)d_in[6];
    const float* Wq1 = (const float*)d_in[7];
    const float* bq1 = (const float*)d_in[8];
    const float* Wk1 = (const float*)d_in[9];
    const float* bk1 = (const float*)d_in[10];
    const float* Wv1 = (const float*)d_in[11];
    const float* bv1 = (const float*)d_in[12];
    const float* Wq2 = (const float*)d_in[13];
    const float* bq2 = (const float*)d_in[14];
    const float* Wk2 = (const float*)d_in[15];
    const float* bk2 = (const float*)d_in[16];
    const float* Wv2 = (const float*)d_in[17];
    const float* bv2 = (const float*)d_in[18];

    const int N = in_sizes[0] / 128;   // 50000
    const int E = in_sizes[1];         // 800000

    float* out = (float*)d_out;        // [N, 384] row-major

    // Workspace layout (bytes):
    //   [0,        N*512*4)  : H1 (encoder hidden) -- reused as q/k/v after encoder
    //   [N*512*4, +E*8*4)    : per-edge score / exp buffer
    //   then m[N*8], s[N*8]
    char*  ws       = (char*)d_ws;
    float* H1       = (float*)ws;
    float* q        = H1;                            // overlay (H1 dead after GEMM2)
    float* kbuf     = H1 + (size_t)N * H_DIM;
    float* vbuf     = H1 + (size_t)2 * N * H_DIM;
    float* scoreBuf = (float*)(ws + (size_t)N * 512 * sizeof(float));
    float* mbuf     = scoreBuf + (size_t)E * NUM_H;
    float* sbuf     = mbuf + (size_t)N * NUM_H;

    // Encoder: z1 = relu(relu(obs @ W1 + b1) @ W2 + b2), z1 -> out[:, 0:128]
    launch_gemm(obs, 128, W1, 512, b1, H1, 512, N, 512, 128, 1, stream);
    launch_gemm(H1, 512, W2, 128, b2, out, OUT_LD, N, 128, 512, 1, stream);

    // GAT layer 1: reads z1 from out cols [0,128), writes z2 to out cols [128,256)
    run_gat_layer(out + 0, OUT_LD, Wq1, bq1, Wk1, bk1, Wv1, bv1,
                  out, 128, q, kbuf, vbuf, scoreBuf, mbuf, sbuf,
                  src, dst, N, E, stream);

    // GAT layer 2: reads z2 from out cols [128,256), writes z3 to out cols [256,384)
    run_gat_layer(out + 128, OUT_LD, Wq2, bq2, Wk2, bk2, Wv2, bv2,
                  out, 256, q, kbuf, vbuf, scoreBuf, mbuf, sbuf,
                  src, dst, N, E, stream);
}